// IdealLearner_60799557042819
// MI455X (gfx1250) — compile-verified
//
#include <hip/hip_runtime.h>
#include <stdint.h>

#define NALGOS 64
#define NTASKS 256
#define TSTEPS 4096

typedef uint32_t u32x4 __attribute__((ext_vector_type(4)));
typedef int32_t  i32x4 __attribute__((ext_vector_type(4)));
typedef int32_t  i32x8 __attribute__((ext_vector_type(8)));

// One block per algorithm 'a' (64 blocks), one thread per task column 'n' (256 = 8 wave32).
// Thread n keeps res[a,n] in a register across all 4096 steps.
// Only the thread with n == lx[t+1] publishes the next uniform coefficient
// coef_{t+1} = eff + boost * sig_t[a, lx[t+1]] into a double-buffered LDS slot.
// One split barrier per step; output store + tm-row prefetch hide in signal->wait window.
__global__ __launch_bounds__(NTASKS, 1)
void ideal_learner_scan(const int* __restrict__ lx,
                        const float* __restrict__ task_matrix,
                        const float* __restrict__ task_difficulty,
                        const float* __restrict__ alg_eff,
                        const float* __restrict__ alg_mem,
                        const float* __restrict__ alg_boost,
                        float* __restrict__ out)
{
    __shared__ int   s_lx[TSTEPS];   // 16 KB step-index table, filled by TDM
    __shared__ float s_coef[2];      // double-buffered per-step uniform coefficient

    const int n = threadIdx.x;       // task column 0..255
    const int a = blockIdx.x;        // algorithm   0..63

    const float eff   = alg_eff[a];
    const float memh  = alg_mem[a];
    const float boost = alg_boost[a];
    const float invd  = 1.0f / task_difficulty[n];

    // ---- Stage the lx table into LDS with the Tensor Data Mover (wave 0 only) ----
    if (threadIdx.x < 32) {
        uint64_t ga      = (uint64_t)(uintptr_t)lx;
        // low 32 bits of the flat shared-aperture address == LDS byte offset (ISA 10.2)
        uint32_t lds_off = (uint32_t)(uintptr_t)(&s_lx[0]);

        u32x4 g0;
        g0[0] = 1u;                                            // count=1, user descriptor
        g0[1] = lds_off;                                       // lds_addr
        g0[2] = (uint32_t)(ga & 0xffffffffu);                  // global_addr[31:0]
        g0[3] = (uint32_t)((ga >> 32) & 0x01ffffffu)           // global_addr[56:32]
              | (2u << 30);                                    // type = 2 ("image")

        i32x8 g1;
        g1[0] = (int)(2u << 16);                               // data_size=2 -> 4B; wg_mask=0
        g1[1] = (int)((TSTEPS & 0xffffu) << 16);               // tensor_dim0 lo16 = 4096
        g1[2] = (int)(((uint32_t)TSTEPS >> 16) | (1u << 16));  // tensor_dim0 hi16 | tensor_dim1=1
        g1[3] = (int)((TSTEPS & 0xffffu) << 16);               // tile_dim0 = 4096
        g1[4] = 1;                                             // tile_dim1=1, tile_dim2=0
        g1[5] = TSTEPS;                                        // tensor_dim0_stride lo32
        g1[6] = (int)((TSTEPS & 0xffffu) << 16);               // stride0 hi16 | stride1 lo16
        g1[7] = 0;                                             // stride1 hi
        i32x4 gz = {0, 0, 0, 0};
#if __clang_major__ >= 23
        i32x8 gz8 = {0, 0, 0, 0, 0, 0, 0, 0};
        __builtin_amdgcn_tensor_load_to_lds(g0, g1, gz, gz, gz8, 0);
#else
        __builtin_amdgcn_tensor_load_to_lds(g0, g1, gz, gz, 0);
#endif
        __builtin_amdgcn_s_wait_tensorcnt(0);
    }
    if (threadIdx.x == 0) s_coef[0] = eff;   // step 0: sig_prev == 0 -> coef = eff
    __syncthreads();

    const int base = (a * NTASKS + n) * (TSTEPS + 1);
    out[base] = 0.0f;                        // out[:,:,0] = z

    // 2-deep software pipeline on task_matrix rows (L2 / WGP$ resident, 256 KB total)
    int   r1     = s_lx[1];                  // lx[t+1]
    int   r2     = s_lx[2];                  // lx[t+2]
    float tm_cur = task_matrix[s_lx[0] * NTASKS + n];
    float tm_nxt = task_matrix[r1 * NTASKS + n];
    float res    = 0.0f;

    for (int t = 0; t < TSTEPS; ++t) {
        float tm_n2 = task_matrix[r2 * NTASKS + n];   // prefetch row for t+2

        float coef = s_coef[t & 1];                   // written by the previous step, fenced by barrier
        res = fmaf(res, memh, tm_cur * coef);         // res*mem + tm*(eff + sp*boost)

        // sig = 2*sigmoid(res/diff) - 1 == sign(y)*(1-e^{-|y|})/(1+e^{-|y|})  (overflow-safe)
        float y   = res * invd;
        float e   = __expf(-fabsf(y));
        float mag = __fdividef(1.0f - e, 1.0f + e);
        float sig = copysignf(mag, y);

        // publish next step's uniform coefficient (exactly one thread matches; at t=T-1 its slot is never read)
        if (n == r1) s_coef[(t + 1) & 1] = fmaf(boost, sig, eff);

        // split barrier: make the publish visible, then hide store+prefetch in the window
        asm volatile("s_wait_dscnt 0\n\ts_barrier_signal -1" ::: "memory");

        out[base + t + 1] = sig;                      // off critical path
        tm_cur = tm_nxt;
        tm_nxt = tm_n2;
        r1 = r2;
        int t3 = t + 3;
        if (t3 > TSTEPS - 1) t3 = TSTEPS - 1;         // clamped tail (last publish is never read)
        r2 = s_lx[t3];

        asm volatile("s_barrier_wait -1" ::: "memory");
    }
}

extern "C" void kernel_launch(void* const* d_in, const int* in_sizes, int n_in,
                              void* d_out, int out_size, void* d_ws, size_t ws_size,
                              hipStream_t stream) {
    const int*   lx    = (const int*)d_in[0];
    const float* tmat  = (const float*)d_in[1];
    const float* diff  = (const float*)d_in[2];
    const float* eff   = (const float*)d_in[3];
    const float* memh  = (const float*)d_in[4];
    const float* boost = (const float*)d_in[5];
    float*       out   = (float*)d_out;

    ideal_learner_scan<<<NALGOS, NTASKS, 0, stream>>>(lx, tmat, diff, eff, memh, boost, out);
}